// PhysicsPDECell_25615184953382
// MI455X (gfx1250) — compile-verified
//
#include <hip/hip_runtime.h>
#include <math.h>

// Problem dims (fixed by the reference).
#define PB 8
#define PT 16
#define PH 512
#define PW 512

constexpr int TILE_W = 64;
constexpr int TILE_H = 32;
constexpr int LDS_W  = 72;          // padded row: [0..2]=unused, 3=left halo, 4..67=body, 68=right halo
constexpr int LDS_H  = TILE_H + 2;  // 34
constexpr int BODY_X = 4;           // body starts at column 4 -> 16B-aligned LDS addresses
constexpr int NTHREADS = 256;       // 8 wave32 waves

// One explicit-Euler PDE step:
//   c_next = c + dt*( -(u*dc/dx + v*dc/dy) + Kd*(1+softplus(pblh))*lap(c) - Ks*rain*c )
// 3x3 correlations (XLA conv = correlation, zero 'SAME' padding):
//   sobel_x = [[-1,0,1],[-2,0,2],[-1,0,1]]/8
//   sobel_y = [[-1,-2,-1],[0,0,0],[1,2,1]]/8
//   lapl    = [[0,1,0],[1,-4,1],[0,1,0]]
__global__ __launch_bounds__(NTHREADS)
void pm25_step_kernel(const float* __restrict__ c_in, long long c_bstride,
                      const float* __restrict__ u,   const float* __restrict__ v,
                      const float* __restrict__ p,   const float* __restrict__ r,
                      const float* __restrict__ dtp, const float* __restrict__ kdp,
                      const float* __restrict__ ksp,
                      float* __restrict__ out)
{
    __shared__ float tile[LDS_H * LDS_W];

    const int b   = blockIdx.z;
    const int tx0 = blockIdx.x * TILE_W;   // multiple of 64 -> body rows 16B aligned
    const int ty0 = blockIdx.y * TILE_H;
    const long long fbase = (long long)b * ((long long)PT * PH * PW); // forcing/out batch offset
    const long long cbase = (long long)b * c_bstride;                 // c batch offset
    const int tid = threadIdx.x;

    // ---- Stage tile body (34 rows x 64 floats) via CDNA5 async global->LDS,
    // 16B-granular: one global_load_async_to_lds_b128 moves 512B per wave issue.
    // Rows outside the image zero-fill with a single ds_store_b128.
    for (int idx = tid; idx < LDS_H * (TILE_W / 4); idx += NTHREADS) {
        const int ly = idx >> 4;        // row 0..33
        const int k  = idx & 15;        // 16B chunk within row
        const int gy = ty0 + ly - 1;
        float* ldst = &tile[ly * LDS_W + BODY_X + 4 * k];
        if (gy >= 0 && gy < PH) {
            const unsigned lds_addr = (unsigned)(size_t)ldst;
            const unsigned long long gaddr =
                (unsigned long long)(const void*)(c_in + cbase + (long long)gy * PW + tx0 + 4 * k);
            asm volatile("global_load_async_to_lds_b128 %0, %1, off"
                         :: "v"(lds_addr), "v"(gaddr)
                         : "memory");
        } else {
            *(float4*)ldst = make_float4(0.0f, 0.0f, 0.0f, 0.0f);
        }
    }
    // ---- Halo edge columns (left gx=tx0-1 -> lx=3, right gx=tx0+64 -> lx=68), b32 each.
    for (int i = tid; i < LDS_H * 2; i += NTHREADS) {
        const int ly   = i >> 1;
        const int side = i & 1;
        const int gy = ty0 + ly - 1;
        const int gx = side ? (tx0 + TILE_W) : (tx0 - 1);
        const int lx = side ? (BODY_X + TILE_W) : (BODY_X - 1);
        float* ldst = &tile[ly * LDS_W + lx];
        if (gy >= 0 && gy < PH && gx >= 0 && gx < PW) {
            const unsigned lds_addr = (unsigned)(size_t)ldst;
            const unsigned long long gaddr =
                (unsigned long long)(const void*)(c_in + cbase + (long long)gy * PW + gx);
            asm volatile("global_load_async_to_lds_b32 %0, %1, off"
                         :: "v"(lds_addr), "v"(gaddr)
                         : "memory");
        } else {
            *ldst = 0.0f;
        }
    }
#if __has_builtin(__builtin_amdgcn_s_wait_asynccnt)
    __builtin_amdgcn_s_wait_asynccnt(0);
#else
    asm volatile("s_wait_asynccnt 0" ::: "memory");
#endif
    __syncthreads();

    const float dt = *dtp;
    const float Kd = *kdp;
    const float Ks = *ksp;

    const int lx  = tid & (TILE_W - 1); // 0..63 : column within tile (coalesced)
    const int ty4 = tid >> 6;           // 0..3  : row phase; each thread does 8 rows
    for (int ly = ty4; ly < TILE_H; ly += 4) {
        const int gy = ty0 + ly;
        const int gx = tx0 + lx;
        const float* Tc = &tile[(ly + 1) * LDS_W + (lx + BODY_X)];

        const float cmm = Tc[-LDS_W - 1], cm0 = Tc[-LDS_W], cmp = Tc[-LDS_W + 1];
        const float c0m = Tc[-1],         c00 = Tc[0],      c0p = Tc[1];
        const float cpm = Tc[ LDS_W - 1], cp0 = Tc[ LDS_W], cpp_ = Tc[ LDS_W + 1];

        const float dcdx = ((cmp - cmm) + 2.0f * (c0p - c0m) + (cpp_ - cpm)) * 0.125f;
        const float dcdy = ((cpm - cmm) + 2.0f * (cp0 - cm0) + (cpp_ - cmp)) * 0.125f;
        const float lap  = cm0 + c0m + c0p + cp0 - 4.0f * c00;

        const long long fi = fbase + (long long)gy * PW + gx;
        const float uu = u[fi];
        const float vv = v[fi];
        const float pb = p[fi];
        const float rn = r[fi];

        // jax.nn.softplus(x) = max(x,0) + log1p(exp(-|x|))  (numerically stable)
        const float sp = fmaxf(pb, 0.0f) + log1pf(expf(-fabsf(pb)));

        const float adv = -(uu * dcdx + vv * dcdy);
        const float dif = Kd * (1.0f + sp) * lap;
        const float scv = -Ks * rn * c00;

        out[fi] = c00 + dt * (adv + dif + scv);
    }
}

extern "C" void kernel_launch(void* const* d_in, const int* in_sizes, int n_in,
                              void* d_out, int out_size, void* d_ws, size_t ws_size,
                              hipStream_t stream)
{
    (void)in_sizes; (void)n_in; (void)d_ws; (void)ws_size; (void)out_size;

    const float* last = (const float*)d_in[0]; // (B,1,H,W)
    const float* u    = (const float*)d_in[1]; // (B,T,H,W)
    const float* v    = (const float*)d_in[2];
    const float* pblh = (const float*)d_in[3];
    const float* rain = (const float*)d_in[4];
    const float* dtp  = (const float*)d_in[5]; // scalars as 1-elem device arrays
    const float* kdp  = (const float*)d_in[6];
    const float* ksp  = (const float*)d_in[7];
    float* out = (float*)d_out;                // (B,T,H,W)

    const long long HW  = (long long)PH * PW;
    const dim3 grid(PW / TILE_W, PH / TILE_H, PB);
    const dim3 block(NTHREADS);

    // 16 sequential steps: step t reads c_{t-1} (last_pm25 for t==0, else the
    // previous output slice) and writes slice t of d_out. Sequential launches
    // on `stream` give the required global ordering; graph-capture safe.
    for (int t = 0; t < PT; ++t) {
        const float* c_in     = (t == 0) ? last : (out + (long long)(t - 1) * HW);
        const long long c_bs  = (t == 0) ? HW : (long long)PT * HW;
        hipLaunchKernelGGL(pm25_step_kernel, grid, block, 0, stream,
                           c_in, c_bs,
                           u + (long long)t * HW, v + (long long)t * HW,
                           pblh + (long long)t * HW, rain + (long long)t * HW,
                           dtp, kdp, ksp,
                           out + (long long)t * HW);
    }
}